// GATNEModel_46952582480106
// MI455X (gfx1250) — compile-verified
//
#include <hip/hip_runtime.h>
#include <math.h>

// Problem constants (fixed by setup_inputs)
#define NUM_NODES 2500
#define T 4
#define EMB 256
#define U_DIM 64
#define DIM_A 32
#define EMBED 512
#define NEIGH 10
#define B_ROWS (NUM_NODES * T)      // 10000
#define NUM_SMS 64
#define MAX_REGION 32
#define NSEG (NUM_SMS * MAX_REGION) // 2048
#define E_TOTAL 131072

typedef __attribute__((ext_vector_type(2))) float v2f;
typedef __attribute__((ext_vector_type(8))) float v8f;

// ---------------------------------------------------------------------------
// Kernel 1: per-row attention aggregation + trans matvec + L2 normalize.
// One 256-thread block per row b. Output: last_node_embed[b][e] (B x 256).
// ---------------------------------------------------------------------------
__global__ void __launch_bounds__(256) k_node_embed(
    const float* __restrict__ node_emb,      // (2500, 256)
    const float* __restrict__ nte_tab,       // (2500, 4, 64)
    const float* __restrict__ tw,            // (4, 64, 256)
    const float* __restrict__ tw_s1,         // (4, 64, 32)
    const float* __restrict__ tw_s2,         // (4, 32)
    const int*   __restrict__ train_inputs,  // (B)
    const int*   __restrict__ train_types,   // (B)
    const int*   __restrict__ node_neigh,    // (B, 4, 10)
    float*       __restrict__ lne)           // (B, 256) out
{
    __shared__ float s_nte[T * U_DIM];   // 256
    __shared__ float s_h[T * DIM_A];     // 128
    __shared__ float s_att[T];
    __shared__ float s_agg[U_DIM];
    __shared__ float s_red[EMB];

    const int b   = blockIdx.x;
    const int tid = threadIdx.x;
    const int typ = train_types[b];

    // node_type_embed[t][u] = sum over 10 neighbors
    {
        const int t = tid >> 6;
        const int u = tid & 63;
        const int* nb = node_neigh + (b * T + t) * NEIGH;
        float acc = 0.f;
#pragma unroll
        for (int n = 0; n < NEIGH; ++n) {
            const int nn = nb[n];
            acc += nte_tab[(nn * T + t) * U_DIM + u];
        }
        s_nte[tid] = acc;
    }
    __syncthreads();

    // h[t][a] = tanh( nte[t][:] . s1[typ][:][a] )
    if (tid < T * DIM_A) {
        const int t = tid >> 5;
        const int a = tid & 31;
        const float* w = tw_s1 + (typ * U_DIM) * DIM_A + a;
        const float* x = s_nte + t * U_DIM;
        float acc = 0.f;
#pragma unroll 8
        for (int u = 0; u < U_DIM; ++u) acc += x[u] * w[u * DIM_A];
        s_h[tid] = tanhf(acc);
    }
    __syncthreads();

    // logits + softmax over T=4 (thread 0)
    if (tid == 0) {
        float logit[T];
        const float* w = tw_s2 + typ * DIM_A;
#pragma unroll
        for (int t = 0; t < T; ++t) {
            float acc = 0.f;
#pragma unroll
            for (int a = 0; a < DIM_A; ++a) acc += s_h[t * DIM_A + a] * w[a];
            logit[t] = acc;
        }
        float mx = logit[0];
#pragma unroll
        for (int t = 1; t < T; ++t) mx = fmaxf(mx, logit[t]);
        float e[T], sum = 0.f;
#pragma unroll
        for (int t = 0; t < T; ++t) { e[t] = expf(logit[t] - mx); sum += e[t]; }
#pragma unroll
        for (int t = 0; t < T; ++t) s_att[t] = e[t] / sum;
    }
    __syncthreads();

    // agg[u] = sum_t att[t] * nte[t][u]
    if (tid < U_DIM) {
        float acc = 0.f;
#pragma unroll
        for (int t = 0; t < T; ++t) acc += s_att[t] * s_nte[t * U_DIM + tid];
        s_agg[tid] = acc;
    }
    __syncthreads();

    // out[e] = node_emb[node][e] + sum_u agg[u] * tw[typ][u][e], then L2 norm
    const int node = train_inputs[b];
    float val = node_emb[node * EMB + tid];
    const float* w = tw + (typ * U_DIM) * EMB + tid;
#pragma unroll 8
    for (int u = 0; u < U_DIM; ++u) val += s_agg[u] * w[u * EMB];

    s_red[tid] = val * val;
    __syncthreads();
#pragma unroll
    for (int s = 128; s > 0; s >>= 1) {
        if (tid < s) s_red[tid] += s_red[tid + s];
        __syncthreads();
    }
    const float scale = 1.0f / fmaxf(sqrtf(s_red[0]), 1e-12f);
    lne[b * EMB + tid] = val * scale;
}

// ---------------------------------------------------------------------------
// Kernel 2: sorted-segment mean + row L2 normalize.
// One 256-thread block per segment. Output: sm_norm (2048 x 256).
// ---------------------------------------------------------------------------
__global__ void __launch_bounds__(256) k_segmean(
    const float* __restrict__ lne,           // (B, 256)
    const int*   __restrict__ region_index,  // (E_TOTAL)
    const int*   __restrict__ seg_ids,       // (E_TOTAL) sorted
    float*       __restrict__ sm_norm)       // (2048, 256) out
{
    __shared__ float s_red[EMB];
    const int seg = blockIdx.x;
    const int tid = threadIdx.x;

    // lower_bound(seg) and upper_bound(seg) in sorted seg_ids
    int lo = 0, hi = E_TOTAL;
    while (lo < hi) { int mid = (lo + hi) >> 1; if (seg_ids[mid] < seg) lo = mid + 1; else hi = mid; }
    const int start = lo;
    hi = E_TOTAL;
    while (lo < hi) { int mid = (lo + hi) >> 1; if (seg_ids[mid] <= seg) lo = mid + 1; else hi = mid; }
    const int end = lo;

    float acc = 0.f;
    for (int i = start; i < end; ++i) {
        const int r = region_index[i];
        acc += lne[r * EMB + tid];
    }
    const int cnt = end - start;
    const float mean = (cnt > 0) ? acc / (float)cnt : 0.f;

    s_red[tid] = mean * mean;
    __syncthreads();
#pragma unroll
    for (int s = 128; s > 0; s >>= 1) {
        if (tid < s) s_red[tid] += s_red[tid + s];
        __syncthreads();
    }
    const float scale = 1.0f / fmaxf(sqrtf(s_red[0]), 1e-12f);
    sm_norm[seg * EMB + tid] = mean * scale;
}

// ---------------------------------------------------------------------------
// Kernel 3: FC GEMM via V_WMMA_F32_16X16X4_F32 + bias + final l2norm.
// Grid: 128 blocks (16 rows each), 256 threads (8 waves). Each wave owns
// 4 N-tiles (n0 = wave*16 + {0,128,256,384}); K = 256 -> 64 WMMAs per tile.
// A layout (f32 16x4): lane<16 -> K={k,k+1}, lane>=16 -> K={k+2,k+3}; M=lane&15.
// B layout mirrors A: lane<16 -> rows K={k,k+1}, lane>=16 -> K={k+2,k+3}; N=lane&15.
// C/D layout: VGPR r -> M = r + 8*(lane>=16), N = lane&15.
// ---------------------------------------------------------------------------
__global__ void __launch_bounds__(256) k_fc_norm(
    const float* __restrict__ sm_norm,  // (2048, 256) = A
    const float* __restrict__ fc_w,     // (512, 256); B[k][n] = fc_w[n][k]
    const float* __restrict__ fc_b,     // (512)
    float*       __restrict__ out)      // (2048, 512)
{
    __shared__ float s_n2[16];

    const int tid  = threadIdx.x;
    const int wave = tid >> 5;
    const int lane = tid & 31;
    const int half = lane >> 4;
    const int l    = lane & 15;
    const int m0   = blockIdx.x * 16;
    const int n0   = wave * 16;

    if (tid < 16) s_n2[tid] = 0.f;
    __syncthreads();

    v8f acc0 = {}, acc1 = {}, acc2 = {}, acc3 = {};

    const float* arow = sm_norm + (m0 + l) * EMB + 2 * half;
    const float* b0r  = fc_w + (n0 + l) * EMB + 2 * half;         // N-tile n0
    const float* b1r  = b0r + 128 * EMB;                          // n0+128
    const float* b2r  = b0r + 256 * EMB;                          // n0+256
    const float* b3r  = b0r + 384 * EMB;                          // n0+384

    for (int k = 0; k < EMB; k += 4) {
        const v2f a = *(const v2f*)(arow + k);
        v2f bb;
        bb = *(const v2f*)(b0r + k);
        acc0 = __builtin_amdgcn_wmma_f32_16x16x4_f32(false, a, false, bb, (short)0, acc0, false, false);
        bb = *(const v2f*)(b1r + k);
        acc1 = __builtin_amdgcn_wmma_f32_16x16x4_f32(false, a, false, bb, (short)0, acc1, false, false);
        bb = *(const v2f*)(b2r + k);
        acc2 = __builtin_amdgcn_wmma_f32_16x16x4_f32(false, a, false, bb, (short)0, acc2, false, false);
        bb = *(const v2f*)(b3r + k);
        acc3 = __builtin_amdgcn_wmma_f32_16x16x4_f32(false, a, false, bb, (short)0, acc3, false, false);
    }

    // bias add + per-row sum of squares (LDS ds_add_f32)
    const float bias0 = fc_b[n0 + l];
    const float bias1 = fc_b[n0 + 128 + l];
    const float bias2 = fc_b[n0 + 256 + l];
    const float bias3 = fc_b[n0 + 384 + l];

    float v0[8], v1[8], v2[8], v3[8];
#pragma unroll
    for (int r = 0; r < 8; ++r) {
        v0[r] = acc0[r] + bias0;
        v1[r] = acc1[r] + bias1;
        v2[r] = acc2[r] + bias2;
        v3[r] = acc3[r] + bias3;
    }
#pragma unroll
    for (int r = 0; r < 8; ++r) {
        const int M = r + 8 * half;
        const float sq = v0[r]*v0[r] + v1[r]*v1[r] + v2[r]*v2[r] + v3[r]*v3[r];
        atomicAdd(&s_n2[M], sq);
    }
    __syncthreads();

    // l2norm denominator: sqrt(sum + 1e-8) + 1e-8 (reference _l2norm)
#pragma unroll
    for (int r = 0; r < 8; ++r) {
        const int M = r + 8 * half;
        const float inv = 1.0f / (sqrtf(s_n2[M] + 1e-8f) + 1e-8f);
        const int row = m0 + M;
        out[row * EMBED + n0 +       l] = v0[r] * inv;
        out[row * EMBED + n0 + 128 + l] = v1[r] * inv;
        out[row * EMBED + n0 + 256 + l] = v2[r] * inv;
        out[row * EMBED + n0 + 384 + l] = v3[r] * inv;
    }
}

// ---------------------------------------------------------------------------
extern "C" void kernel_launch(void* const* d_in, const int* in_sizes, int n_in,
                              void* d_out, int out_size, void* d_ws, size_t ws_size,
                              hipStream_t stream) {
    const float* node_emb     = (const float*)d_in[0];   // (2500,256)
    const float* nte_tab      = (const float*)d_in[1];   // (2500,4,64)
    const float* tw           = (const float*)d_in[2];   // (4,64,256)
    const float* tw_s1        = (const float*)d_in[3];   // (4,64,32)
    const float* tw_s2        = (const float*)d_in[4];   // (4,32,1)
    const float* fc_w         = (const float*)d_in[5];   // (512,256)
    const float* fc_b         = (const float*)d_in[6];   // (512)
    const int*   train_inputs = (const int*)d_in[7];     // (B)
    const int*   train_types  = (const int*)d_in[8];     // (B)
    const int*   node_neigh   = (const int*)d_in[9];     // (B,4,10)
    const int*   region_index = (const int*)d_in[10];    // (131072)
    const int*   seg_ids      = (const int*)d_in[11];    // (131072) sorted

    float* lne     = (float*)d_ws;                        // B x 256
    float* sm_norm = lne + (size_t)B_ROWS * EMB;          // 2048 x 256
    float* outp    = (float*)d_out;                       // 2048 x 512

    k_node_embed<<<B_ROWS, 256, 0, stream>>>(node_emb, nte_tab, tw, tw_s1, tw_s2,
                                             train_inputs, train_types, node_neigh, lne);
    k_segmean<<<NSEG, 256, 0, stream>>>(lne, region_index, seg_ids, sm_norm);
    k_fc_norm<<<NSEG / 16, 256, 0, stream>>>(sm_norm, fc_w, fc_b, outp);
}